// HyperbolicAggregation_56341380989449
// MI455X (gfx1250) — compile-verified
//
#include <hip/hip_runtime.h>
#include <math.h>

// ---------------------------------------------------------------------------
// HyperbolicAggregation for MI455X (gfx1250, wave32)
//   out = proj(expmap0(adj @ logmap0(x)))
// adj: 16384x16384 f32 (1 GiB, streamed once -> HBM-bound, ~46us floor at
// 23.3 TB/s). Matmul runs on V_WMMA_F32_16X16X4_F32 (fp32 kept: adj entries
// ~6e-5 after row-normalization; we're bandwidth-bound so full precision is
// free). Tangent matrix is stored transposed (D-major) so both A and B
// fragments are single global_load_b64 per lane per WMMA.
// ---------------------------------------------------------------------------

typedef __attribute__((ext_vector_type(2))) float v2f;
typedef __attribute__((ext_vector_type(4))) float v4f;
typedef __attribute__((ext_vector_type(8))) float v8f;

#define DDIM 16
#define EPS_F 1e-15f
#define MAX_NORM_F 0.99999f        /* (1 - 1e-5) / sqrt(c), c = 1 */
#define ATANH_CLIP_F 0.9999999f    /* 1 - 1e-7 */

// XOR-butterfly add across 16-lane halves via ds_swizzle (group-of-32 mode:
// imm = xor_mask<<10 | or_mask<<5 | and_mask; and_mask=0x1f keeps all lanes).
// ds_swizzle immediates must be literal constants -> hand-unrolled.
__device__ __forceinline__ float swz_add(float v, int sw_imm_result) {
    return v + __int_as_float(sw_imm_result);
}

__device__ __forceinline__ float lane_xor_add16(float v) {
    v = swz_add(v, __builtin_amdgcn_ds_swizzle(__float_as_int(v),
                                               (1 << 10) | 0x1f));
    v = swz_add(v, __builtin_amdgcn_ds_swizzle(__float_as_int(v),
                                               (2 << 10) | 0x1f));
    v = swz_add(v, __builtin_amdgcn_ds_swizzle(__float_as_int(v),
                                               (4 << 10) | 0x1f));
    v = swz_add(v, __builtin_amdgcn_ds_swizzle(__float_as_int(v),
                                               (8 << 10) | 0x1f));
    return v;
}

// ---------------------------------------------------------------------------
// Kernel 1: x_tangent = logmap0(x), stored TRANSPOSED: xtT[d][i], d=0..15.
// One thread per row; stores for fixed d are lane-contiguous (coalesced).
// ---------------------------------------------------------------------------
__global__ void hyp_logmap0_t_kernel(const float* __restrict__ x,
                                     float* __restrict__ xtT, int n) {
    int i = blockIdx.x * blockDim.x + threadIdx.x;
    if (i >= n) return;
    const v4f* row = (const v4f*)(x + (size_t)i * DDIM);
    v4f r0 = row[0], r1 = row[1], r2 = row[2], r3 = row[3];
    float ss = 0.f;
#pragma unroll
    for (int j = 0; j < 4; ++j) {
        ss += r0[j] * r0[j];
        ss += r1[j] * r1[j];
        ss += r2[j] * r2[j];
        ss += r3[j] * r3[j];
    }
    float nrm = fmaxf(sqrtf(ss), EPS_F);
    float arg = fminf(nrm, ATANH_CLIP_F);
    float s = atanhf(arg) / nrm;
    v4f t0 = r0 * s, t1 = r1 * s, t2 = r2 * s, t3 = r3 * s;
#pragma unroll
    for (int j = 0; j < 4; ++j) {
        xtT[(size_t)(j)      * n + i] = t0[j];
        xtT[(size_t)(j + 4)  * n + i] = t1[j];
        xtT[(size_t)(j + 8)  * n + i] = t2[j];
        xtT[(size_t)(j + 12) * n + i] = t3[j];
    }
}

// ---------------------------------------------------------------------------
// Kernel 2: one wave32 per 16-row output tile, K-chunk = 64 (16 WMMAs/iter).
// A layout (ISA 7.12.2, 32-bit A 16x4): lane L<16 -> (M=L, K=0..1),
// lane L+16 -> (M=L, K=2..3)  => one b64 per lane per WMMA. B mirrors it
// from the transposed tangent matrix (also one b64 per lane per WMMA).
// C/D layout: vgpr r = row r (lanes 0-15) and row r+8 (lanes 16-31).
// ---------------------------------------------------------------------------
__global__ __launch_bounds__(32) void hyp_agg_wmma_kernel(
    const float* __restrict__ adj, const float* __restrict__ xtT,
    float* __restrict__ out, int n) {
    const int lane  = threadIdx.x;        // 0..31
    const int mrow  = lane & 15;          // A row within tile / B column
    const int khalf = (lane >> 4) * 2;    // 0 or 2: K sub-offset per lane half
    const int rowBase = blockIdx.x * 16;

    const float* aRow = adj + (size_t)(rowBase + mrow) * (size_t)n + khalf;
    const float* bRow = xtT + (size_t)mrow * (size_t)n + khalf;
    // Lanes 16-31 prefetch the next 128B line so one prefetch op covers the
    // full 256B/row consumed per iteration, 8KB (32 iters) ahead of demand.
    const float* pfRow = aRow + 2048 + (lane >> 4) * 32;

    v8f acc = {};

    for (int k = 0; k < n; k += 64) {
        if (k + 2112 < n) {
            __builtin_prefetch(pfRow + k, 0, 1);   // stream into far cache
        }
        v2f A[16];
        v2f B[16];
#pragma unroll
        for (int t = 0; t < 16; ++t) {
            const int kk = k + 4 * t;
            A[t] = *(const v2f*)(aRow + kk);       // b64, 8B aligned
            B[t] = *(const v2f*)(bRow + kk);       // b64, 8B aligned
        }
#pragma unroll
        for (int t = 0; t < 16; ++t) {
            // (neg_a, A, neg_b, B, c_mod, C, reuse_a, reuse_b)
            acc = __builtin_amdgcn_wmma_f32_16x16x4_f32(
                false, A[t], false, B[t], (short)0, acc, false, false);
        }
    }

    // Epilogue: fused expmap0 + proj. acc[r] holds one row's 16 values across
    // a 16-lane half -> ds_swizzle butterfly for the row norm.
    //   u -> u * min(tanh(|u|), MAX_NORM) / max(|u|, eps)
#pragma unroll
    for (int r = 0; r < 8; ++r) {
        float v = acc[r];
        float ss = lane_xor_add16(v * v);
        float nrm = fmaxf(sqrtf(ss), EPS_F);
        float f = fminf(tanhf(nrm), MAX_NORM_F) / nrm;
        int orow = rowBase + r + (lane >> 4) * 8;
        out[(size_t)orow * DDIM + mrow] = v * f;
    }
}

// ---------------------------------------------------------------------------
extern "C" void kernel_launch(void* const* d_in, const int* in_sizes, int n_in,
                              void* d_out, int out_size, void* d_ws,
                              size_t ws_size, hipStream_t stream) {
    const float* x   = (const float*)d_in[0];   // [N, 16] f32
    const float* adj = (const float*)d_in[1];   // [N, N]  f32
    float* out = (float*)d_out;                 // [N, 16] f32
    float* xtT = (float*)d_ws;                  // scratch: [16, N] f32 = 1 MB

    const int n = in_sizes[0] / DDIM;           // N = 16384

    hyp_logmap0_t_kernel<<<(n + 255) / 256, 256, 0, stream>>>(x, xtT, n);
    hyp_agg_wmma_kernel<<<n / 16, 32, 0, stream>>>(adj, xtT, out, n);
}